// FastHoloLinear_19301583029007
// MI455X (gfx1250) — compile-verified
//
#include <hip/hip_runtime.h>
#include <math.h>

// FastHoloLinear on MI455X (gfx1250, wave32).
//
// out = (x @ basis^T) @ (amp * cos(phase))^T
//   x [8192,4096] f32, basis [128,4096] f32, phase/amp [4096,128] f32.
//
// Roofline: ~262 MB mandatory HBM traffic at 23.3 TB/s -> ~11 us floor;
// only ~17.2 GFLOP, so both GEMMs are memory-bound even with full-precision
// v_wmma_f32_16x16x4_f32 -> keep fp32 end to end (no bf16 downconvert).
//
// CDNA5 async data path:
//  - GLOBAL_LOAD_ASYNC_TO_LDS_B128 (ASYNCcnt) with LDS double-buffering so
//    HBM loads overlap WMMA compute, no VGPR round-trip for staging.
//  - non-temporal stores for the 128 MB output stream (keeps R/W hot in L2).

typedef __attribute__((ext_vector_type(2))) float v2f;
typedef __attribute__((ext_vector_type(4))) float v4f;
typedef __attribute__((ext_vector_type(8))) float v8f;
typedef __attribute__((ext_vector_type(4))) int   v4i;

#define IN_F  4096
#define OUT_F 4096
#define HARM  128

#if defined(__HIP_DEVICE_COMPILE__) && \
    __has_builtin(__builtin_amdgcn_global_load_async_to_lds_b128) && \
    __has_builtin(__builtin_amdgcn_s_wait_asynccnt)
#define HAVE_ASYNC_LDS 1
#endif

#ifdef HAVE_ASYNC_LDS
typedef __attribute__((address_space(1))) v4i* gp_v4i;
typedef __attribute__((address_space(3))) v4i* lp_v4i;
#endif

// Copy 16 bytes global -> LDS. Async (no VGPR round-trip) when available.
__device__ __forceinline__ void cp16(const float* __restrict__ g, float* l) {
#ifdef HAVE_ASYNC_LDS
    __builtin_amdgcn_global_load_async_to_lds_b128(
        (gp_v4i)(g), (lp_v4i)(l), /*offset=*/0, /*cpol=*/0);
#else
    *(v4f*)l = *(const v4f*)g;
#endif
}

#ifdef HAVE_ASYNC_LDS
#define ASYNC_WAIT(n) __builtin_amdgcn_s_wait_asynccnt(n)
#else
#define ASYNC_WAIT(n)
#endif

// ---------------------------------------------------------------------------
// Kernel 1: W[o,h] = amp[o,h] * cos(phase[o,h])
// ---------------------------------------------------------------------------
__global__ __launch_bounds__(256) void wprep_kernel(const float* __restrict__ phase,
                                                    const float* __restrict__ amp,
                                                    float* __restrict__ W, int n) {
    int i = blockIdx.x * blockDim.x + threadIdx.x;
    if (i < n) W[i] = amp[i] * cosf(phase[i]);
}

// ---------------------------------------------------------------------------
// Kernel 2: R[M,128] = X[M,4096] @ basis[128,4096]^T
// 128 threads = 4 waves; block tile M=64 x N=128; K chunks of 32,
// double-buffered through LDS via async loads.
// LDS row stride 36 floats: 36*lo mod 64 distinct for lo=0..15, +2*hi covers
// all 64 banks exactly once -> conflict-free fragment reads; 144 B row pitch
// keeps b128 writes 16B-aligned.
// ---------------------------------------------------------------------------
#define G1_KC   32
#define G1_LS   36
#define G1_SXN  (64 * G1_LS)
#define G1_SBN  (128 * G1_LS)
#define G1_NASY 12            // async b128 per thread per chunk (4 X + 8 basis)

__global__ __launch_bounds__(128) void gemm1_kernel(const float* __restrict__ X,
                                                    const float* __restrict__ Bs,
                                                    float* __restrict__ R, int M) {
    __shared__ float sx[2 * G1_SXN];   // 18.4 KB
    __shared__ float sb[2 * G1_SBN];   // 36.9 KB

    const int t    = threadIdx.x;
    const int wave = t >> 5;
    const int lane = t & 31;
    const int lo   = lane & 15;
    const int hi   = lane >> 4;
    const int row0 = blockIdx.x * 64;

    v8f acc[8];
#pragma unroll
    for (int i = 0; i < 8; ++i) acc[i] = {};

    auto issue = [&](int kc, int buf) {
#pragma unroll
        for (int i = 0; i < 4; ++i) {            // X tile: 64x32 = 512 x b128
            int idx = t + i * 128;
            int r = idx >> 3, c4 = idx & 7;
            cp16(X + (size_t)(row0 + r) * IN_F + kc + c4 * 4,
                 sx + buf * G1_SXN + r * G1_LS + c4 * 4);
        }
#pragma unroll
        for (int i = 0; i < 8; ++i) {            // basis tile: 128x32 = 1024 x b128
            int idx = t + i * 128;
            int r = idx >> 3, c4 = idx & 7;
            cp16(Bs + (size_t)r * IN_F + kc + c4 * 4,
                 sb + buf * G1_SBN + r * G1_LS + c4 * 4);
        }
    };

    issue(0, 0);
    const int NCH = IN_F / G1_KC;                // 128 chunks
    for (int c = 0; c < NCH; ++c) {
        const int cur = c & 1;
        if (c + 1 < NCH) {
            issue((c + 1) * G1_KC, cur ^ 1);     // prefetch next chunk
            ASYNC_WAIT(G1_NASY);                 // chunk c done, c+1 in flight
        } else {
            ASYNC_WAIT(0);
        }
        __syncthreads();                          // chunk c visible to all waves

        const float* ax = sx + cur * G1_SXN + (wave * 16 + lo) * G1_LS + 2 * hi;
        const float* bx = sb + cur * G1_SBN + lo * G1_LS + 2 * hi;
#pragma unroll
        for (int k = 0; k < G1_KC; k += 4) {
            v2f a = *(const v2f*)(ax + k);
#pragma unroll
            for (int nt = 0; nt < 8; ++nt) {
                v2f b = *(const v2f*)(bx + nt * 16 * G1_LS + k);
                acc[nt] = __builtin_amdgcn_wmma_f32_16x16x4_f32(
                    false, a, false, b, (short)0, acc[nt], false, false);
            }
        }
        __syncthreads();                          // all waves done before overwrite
    }

    // R stays hot in L2 for gemm2 -> regular (RT) stores.
#pragma unroll
    for (int nt = 0; nt < 8; ++nt)
#pragma unroll
        for (int r = 0; r < 8; ++r) {
            int row = row0 + wave * 16 + r + hi * 8;
            int col = nt * 16 + lo;
            R[(size_t)row * HARM + col] = acc[nt][r];
        }
}

// ---------------------------------------------------------------------------
// Kernel 3: OUT[M,4096] = R[M,128] @ W[4096,128]^T
// 256 threads = 8 waves; block tile M=128 x N=64; K=128 in 4 chunks of 32,
// double-buffered async. Output stored non-temporally (write-once stream).
// ---------------------------------------------------------------------------
#define G2_KC   32
#define G2_LS   36
#define G2_SRN  (128 * G2_LS)
#define G2_SWN  (64 * G2_LS)
#define G2_NASY 6             // async b128 per thread per chunk (4 R + 2 W)

__global__ __launch_bounds__(256) void gemm2_kernel(const float* __restrict__ R,
                                                    const float* __restrict__ W,
                                                    float* __restrict__ OUT, int M) {
    __shared__ float sr[2 * G2_SRN];   // 36.9 KB
    __shared__ float sw[2 * G2_SWN];   // 18.4 KB

    const int t    = threadIdx.x;
    const int wave = t >> 5;
    const int lane = t & 31;
    const int lo   = lane & 15;
    const int hi   = lane >> 4;
    const int row0 = blockIdx.x * 128;
    const int col0 = blockIdx.y * 64;

    v8f acc[4];
#pragma unroll
    for (int i = 0; i < 4; ++i) acc[i] = {};

    auto issue = [&](int kc, int buf) {
#pragma unroll
        for (int i = 0; i < 4; ++i) {            // R tile: 128x32 = 1024 x b128
            int idx = t + i * 256;
            int r = idx >> 3, c4 = idx & 7;
            cp16(R + (size_t)(row0 + r) * HARM + kc + c4 * 4,
                 sr + buf * G2_SRN + r * G2_LS + c4 * 4);
        }
#pragma unroll
        for (int i = 0; i < 2; ++i) {            // W tile: 64x32 = 512 x b128
            int idx = t + i * 256;
            int r = idx >> 3, c4 = idx & 7;
            cp16(W + (size_t)(col0 + r) * HARM + kc + c4 * 4,
                 sw + buf * G2_SWN + r * G2_LS + c4 * 4);
        }
    };

    issue(0, 0);
    const int NCH = HARM / G2_KC;                // 4 chunks
    for (int c = 0; c < NCH; ++c) {
        const int cur = c & 1;
        if (c + 1 < NCH) {
            issue((c + 1) * G2_KC, cur ^ 1);
            ASYNC_WAIT(G2_NASY);
        } else {
            ASYNC_WAIT(0);
        }
        __syncthreads();

        const float* ar = sr + cur * G2_SRN + (wave * 16 + lo) * G2_LS + 2 * hi;
        const float* bw = sw + cur * G2_SWN + lo * G2_LS + 2 * hi;
#pragma unroll
        for (int k = 0; k < G2_KC; k += 4) {
            v2f a = *(const v2f*)(ar + k);
#pragma unroll
            for (int nt = 0; nt < 4; ++nt) {
                v2f b = *(const v2f*)(bw + nt * 16 * G2_LS + k);
                acc[nt] = __builtin_amdgcn_wmma_f32_16x16x4_f32(
                    false, a, false, b, (short)0, acc[nt], false, false);
            }
        }
        __syncthreads();
    }

    // 128 MB write-once stream: non-temporal to avoid evicting R/W from L2.
#pragma unroll
    for (int nt = 0; nt < 4; ++nt)
#pragma unroll
        for (int r = 0; r < 8; ++r) {
            int row = row0 + wave * 16 + r + hi * 8;
            int col = col0 + nt * 16 + lo;
            __builtin_nontemporal_store(acc[nt][r], OUT + (size_t)row * OUT_F + col);
        }
}

// ---------------------------------------------------------------------------
extern "C" void kernel_launch(void* const* d_in, const int* in_sizes, int n_in,
                              void* d_out, int out_size, void* d_ws, size_t ws_size,
                              hipStream_t stream) {
    const float* x     = (const float*)d_in[0];
    const float* basis = (const float*)d_in[1];
    const float* phase = (const float*)d_in[2];
    const float* amp   = (const float*)d_in[3];
    float*       out   = (float*)d_out;

    const int M = in_sizes[0] / IN_F;            // 8192

    // Workspace: W [OUT_F*HARM] then R [M*HARM]  (~6 MB total)
    float* W = (float*)d_ws;
    float* R = W + (size_t)OUT_F * HARM;

    const int nW = OUT_F * HARM;
    wprep_kernel<<<(nW + 255) / 256, 256, 0, stream>>>(phase, amp, W, nW);

    gemm1_kernel<<<M / 64, 128, 0, stream>>>(x, basis, R, M);

    dim3 g2(M / 128, OUT_F / 64);
    gemm2_kernel<<<g2, 256, 0, stream>>>(R, W, out, M);
}